// Linear_DS_agent_26886495273774
// MI455X (gfx1250) — compile-verified
//
#include <hip/hip_runtime.h>
#include <math.h>

typedef __attribute__((ext_vector_type(2))) float v2f;
typedef __attribute__((ext_vector_type(4))) float v4f;
typedef __attribute__((ext_vector_type(8))) float v8f;

#define T_STEP   0.01f
#define STD_F    0.01f
#define MAX_U    15000.0f
#define N_STEPS  2048
#define N_ARMS   5000
#define N_CHAINS 16
#define CHAIN_LEN 128   /* N_STEPS / N_CHAINS */
#define HS       10

#if defined(__has_builtin) && __has_builtin(__builtin_amdgcn_tanhf)
__device__ __forceinline__ float fast_tanh(float x) { return __builtin_amdgcn_tanhf(x); }
#else
__device__ __forceinline__ float fast_tanh(float x) { return tanhf(x); }
#endif

// ---------------------------------------------------------------------------
// Kernel 1: single wave (32 lanes). MLP -> x0, invert P, build M = I - dt*A,
// M^128 via squaring, 16 chain-start states, then 128 iterations where both
// the action projection (Wm_pad @ X) and the state update (M @ X) are done
// with V_WMMA_F32_16X16X4_F32 (4 WMMAs each, K=4). Writes actions[2048*2].
// ---------------------------------------------------------------------------
__global__ __launch_bounds__(32)
void rollout_wmma_kernel(const float* __restrict__ target,
                         const float* __restrict__ D,
                         const float* __restrict__ P,
                         const float* __restrict__ W1,
                         const float* __restrict__ b1,
                         const float* __restrict__ W2,
                         const float* __restrict__ b2,
                         const float* __restrict__ Wm,
                         const float* __restrict__ bm,
                         float* __restrict__ actions /* [N_STEPS*2] */)
{
    __shared__ float h[256];
    __shared__ float aug[10][20];
    __shared__ float invP[10][10];
    __shared__ float Mmat[16][16];   // M = I - dt*A, zero padded
    __shared__ float Mp[16][16];     // M^128
    __shared__ float Tmp[16][16];
    __shared__ float WmS[16][16];    // Wm padded (rows 0..1)
    __shared__ float Xs[16][16];     // state matrix, row-major (row=state dim, col=chain)
    __shared__ float xvec[16];

    const int lane = threadIdx.x;   // 0..31
    const int hi   = lane >> 4;     // lane half
    const int lo   = lane & 15;

    // ---- h = relu(W1 @ target + b1) ----
    const float t0 = target[0], t1 = target[1];
    for (int k = 0; k < 8; ++k) {
        int i = lane + 32 * k;
        float v = W1[i * 2 + 0] * t0 + W1[i * 2 + 1] * t1 + b1[i];
        h[i] = v > 0.f ? v : 0.f;
    }
    // zero-pad the matrices while waiting
    for (int k = 0; k < 8; ++k) {
        int e = lane + 32 * k;
        ((float*)Mmat)[e] = 0.f;
        ((float*)WmS)[e]  = 0.f;
    }
    __syncthreads();

    // ---- x0 = W2 @ h + b2 (rows 10..15 zero padded) ----
    if (lane < 16) {
        float acc = 0.f;
        if (lane < HS) {
            acc = b2[lane];
            for (int j = 0; j < 256; ++j) acc += W2[lane * 256 + j] * h[j];
        }
        xvec[lane] = acc;
    }
    __syncthreads();

    // ---- invP = inv(P): Gauss-Jordan with partial pivoting, serial lane 0 ----
    if (lane == 0) {
        for (int i = 0; i < 10; ++i)
            for (int j = 0; j < 10; ++j) {
                aug[i][j]      = P[i * 10 + j];
                aug[i][10 + j] = (i == j) ? 1.f : 0.f;
            }
        for (int c = 0; c < 10; ++c) {
            int piv = c;
            for (int r = c + 1; r < 10; ++r)
                if (fabsf(aug[r][c]) > fabsf(aug[piv][c])) piv = r;
            if (piv != c)
                for (int j = 0; j < 20; ++j) {
                    float sw = aug[c][j]; aug[c][j] = aug[piv][j]; aug[piv][j] = sw;
                }
            float inv = 1.f / aug[c][c];
            for (int j = 0; j < 20; ++j) aug[c][j] *= inv;
            for (int r = 0; r < 10; ++r) if (r != c) {
                float f = aug[r][c];
                for (int j = 0; j < 20; ++j) aug[r][j] -= f * aug[c][j];
            }
        }
        for (int i = 0; i < 10; ++i)
            for (int j = 0; j < 10; ++j) invP[i][j] = aug[i][10 + j];
    }
    __syncthreads();

    // ---- M = I - dt * (P * exp(D)) @ invP ----
    for (int e = lane; e < 100; e += 32) {
        int i = e / 10, j = e % 10;
        float a = 0.f;
        for (int k = 0; k < 10; ++k)
            a += P[i * 10 + k] * expf(D[k]) * invP[k][j];
        Mmat[i][j] = ((i == j) ? 1.f : 0.f) - T_STEP * a;
    }
    if (lane < 20) WmS[lane / 10][lane % 10] = Wm[lane];
    __syncthreads();

    // ---- Mp = M^128 via 7 squarings (10x10 live region) ----
    for (int e = lane; e < 256; e += 32) ((float*)Mp)[e] = ((float*)Mmat)[e];
    __syncthreads();
    for (int s = 0; s < 7; ++s) {
        for (int e = lane; e < 100; e += 32) {
            int i = e / 10, j = e % 10;
            float a = 0.f;
            for (int k = 0; k < 10; ++k) a += Mp[i][k] * Mp[k][j];
            Tmp[i][j] = a;
        }
        __syncthreads();
        for (int e = lane; e < 100; e += 32) Mp[e / 10][e % 10] = Tmp[e / 10][e % 10];
        __syncthreads();
    }

    // ---- chain starts: Xs[:,c] = (M^128)^c @ x0 ----
    for (int e = lane; e < 256; e += 32) ((float*)Xs)[e] = 0.f;
    __syncthreads();
    for (int c = 0; c < N_CHAINS; ++c) {
        if (lane < 16) Xs[lane][c] = xvec[lane];
        __syncthreads();
        float nv = 0.f;
        if (lane < HS)
            for (int k = 0; k < HS; ++k) nv += Mp[lane][k] * xvec[k];
        __syncthreads();
        if (lane < 16) xvec[lane] = (lane < HS) ? nv : 0.f;
        __syncthreads();
    }

    // ---- load A-operand tiles (16x4 f32 layout: lanes0-15 hold K 0,1;
    //      lanes16-31 hold K 2,3 across the 2 VGPRs) ----
    v2f m_a[4], w_a[4];
    for (int k = 0; k < 4; ++k) {
        m_a[k].x = Mmat[lo][4 * k + 2 * hi + 0];
        m_a[k].y = Mmat[lo][4 * k + 2 * hi + 1];
        w_a[k].x = WmS [lo][4 * k + 2 * hi + 0];
        w_a[k].y = WmS [lo][4 * k + 2 * hi + 1];
    }
    // C/D layout: VGPR v holds row v (lanes 0-15) / row v+8 (lanes 16-31)
    v8f x_cd;
    for (int v = 0; v < 8; ++v) x_cd[v] = Xs[v + 8 * hi][lo];
    __syncthreads();

    const float bm0 = bm[0], bm1 = bm[1];

    for (int t = 0; t < CHAIN_LEN; ++t) {
        // publish current state row-major for B-operand re-layout
        for (int v = 0; v < 8; ++v) Xs[v + 8 * hi][lo] = x_cd[v];
        __syncthreads();
        // B tiles (4x16): VGPR0/1 hold K {0,1} (lanes 0-15) / {2,3} (lanes 16-31)
        v2f b0, b1v, b2v, b3v;
        b0.x  = Xs[ 0 + 2 * hi][lo];  b0.y  = Xs[ 1 + 2 * hi][lo];
        b1v.x = Xs[ 4 + 2 * hi][lo];  b1v.y = Xs[ 5 + 2 * hi][lo];
        b2v.x = Xs[ 8 + 2 * hi][lo];  b2v.y = Xs[ 9 + 2 * hi][lo];
        b3v.x = Xs[12 + 2 * hi][lo];  b3v.y = Xs[13 + 2 * hi][lo];
        __syncthreads();

        // actions(t) = tanh(Wm_pad @ X + bm): rows 0,1 of Y are valid
        v8f yacc = {};
        yacc = __builtin_amdgcn_wmma_f32_16x16x4_f32(false, w_a[0], false, b0,  (short)0, yacc, false, false);
        yacc = __builtin_amdgcn_wmma_f32_16x16x4_f32(false, w_a[1], false, b1v, (short)0, yacc, false, false);
        yacc = __builtin_amdgcn_wmma_f32_16x16x4_f32(false, w_a[2], false, b2v, (short)0, yacc, false, false);
        yacc = __builtin_amdgcn_wmma_f32_16x16x4_f32(false, w_a[3], false, b3v, (short)0, yacc, false, false);

        if (lane < 16) {                       // column n = chain index
            int step = lane * CHAIN_LEN + t;   // global step for this chain
            actions[step * 2 + 0] = fast_tanh(yacc[0] + bm0);
            actions[step * 2 + 1] = fast_tanh(yacc[1] + bm1);
        }

        // X <- M @ X
        v8f xacc = {};
        xacc = __builtin_amdgcn_wmma_f32_16x16x4_f32(false, m_a[0], false, b0,  (short)0, xacc, false, false);
        xacc = __builtin_amdgcn_wmma_f32_16x16x4_f32(false, m_a[1], false, b1v, (short)0, xacc, false, false);
        xacc = __builtin_amdgcn_wmma_f32_16x16x4_f32(false, m_a[2], false, b2v, (short)0, xacc, false, false);
        xacc = __builtin_amdgcn_wmma_f32_16x16x4_f32(false, m_a[3], false, b3v, (short)0, xacc, false, false);
        x_cd = xacc;
    }
}

// ---------------------------------------------------------------------------
// Kernel 2: pure-bandwidth broadcast. out[i] = (act[i % 4096] + 0.01*eps[i])*15000
// The reference's final reshape (arms, 2, steps) is a view: flat layout unchanged.
// eps/out are streamed non-temporally (read/written exactly once); the 16KB
// action table stays cache-resident. 5,120,000 float4 = 20000 blocks x 256 thr.
// ---------------------------------------------------------------------------
__global__ __launch_bounds__(256)
void broadcast_sample_kernel(const v4f* __restrict__ eps4,
                             const v4f* __restrict__ act4,
                             v4f* __restrict__ out4)
{
    const long long i = (long long)blockIdx.x * blockDim.x + threadIdx.x;
    const int j = (int)(i & 1023);       // (N_STEPS*2)/4 = 1024 action vec4s
    v4f e = __builtin_nontemporal_load(&eps4[i]);
    v4f a = act4[j];
    v4f r = (a + STD_F * e) * MAX_U;
    __builtin_nontemporal_store(r, &out4[i]);
}

extern "C" void kernel_launch(void* const* d_in, const int* in_sizes, int n_in,
                              void* d_out, int out_size, void* d_ws, size_t ws_size,
                              hipStream_t stream) {
    const float* target = (const float*)d_in[0];
    const float* eps    = (const float*)d_in[1];
    const float* D      = (const float*)d_in[2];
    const float* P      = (const float*)d_in[3];
    const float* W1     = (const float*)d_in[4];
    const float* b1     = (const float*)d_in[5];
    const float* W2     = (const float*)d_in[6];
    const float* b2     = (const float*)d_in[7];
    const float* Wm     = (const float*)d_in[8];
    const float* bm     = (const float*)d_in[9];

    float* actions = (float*)d_ws;       // 4096 floats
    float* out     = (float*)d_out;      // 20,480,000 floats

    hipLaunchKernelGGL(rollout_wmma_kernel, dim3(1), dim3(32), 0, stream,
                       target, D, P, W1, b1, W2, b2, Wm, bm, actions);

    const long long n4 = (long long)N_ARMS * N_STEPS * 2 / 4;  // 5,120,000
    const int threads = 256;
    const int blocks = (int)(n4 / threads);                     // 20,000 exact
    hipLaunchKernelGGL(broadcast_sample_kernel, dim3(blocks), dim3(threads), 0, stream,
                       (const v4f*)eps, (const v4f*)actions, (v4f*)out);
}